// pcaWhitening_40432822125226
// MI455X (gfx1250) — compile-verified
//
#include <hip/hip_runtime.h>

// Problem constants (from setup_inputs: x=[64,32,128,128] f32, mb=4, bits=8)
#define B_DIM 64
#define C_DIM 32
#define H_DIM 128
#define W_DIM 128
#define MB 4
#define F_DIM 512            // C * mb * mb
#define N_DIM 65536          // (H/mb)*(W/mb)*B
#define TOTAL (B_DIM * C_DIM * H_DIM * W_DIM)   // 33,554,432
#define NSWEEPS 8
#define KSLICES 16           // K-split for the cov GEMM (atomic reduction)

typedef float v2f __attribute__((ext_vector_type(2)));
typedef float v8f __attribute__((ext_vector_type(8)));
typedef unsigned int u32x4 __attribute__((ext_vector_type(4)));
typedef int i32x4 __attribute__((ext_vector_type(4)));
typedef int i32x8 __attribute__((ext_vector_type(8)));

// ---------------------------------------------------------------------------
// D = A(16x4 f32) x B(4x16 f32) + C(16x16 f32), wave32 WMMA
// ---------------------------------------------------------------------------
__device__ __forceinline__ v8f wmma_f32_k4(v2f a, v2f b, v8f c) {
  return __builtin_amdgcn_wmma_f32_16x16x4_f32(
      /*neg_a=*/false, a, /*neg_b=*/false, b,
      /*c_mod=*/(short)0, c, /*reuse_a=*/false, /*reuse_b=*/false);
}

// ---------------------------------------------------------------------------
// TDM: DMA a 64x64 f32 tile (row stride N_DIM elems) from global into LDS,
// padding 1 dword after every 64 dwords -> LDS row stride 65 floats
// (bank-conflict-free column access). D# bitfields per ISA ch.8.
// ---------------------------------------------------------------------------
__device__ __forceinline__ void tdm_load_tile64x64(const float* gtile,
                                                   unsigned ldsByteAddr) {
  unsigned long long ga = (unsigned long long)(uintptr_t)gtile;
  u32x4 g0;
  g0[0] = 1u;                                           // count=1, user mode
  g0[1] = ldsByteAddr;                                  // lds_addr
  g0[2] = (unsigned)(ga & 0xFFFFFFFFu);                 // global_addr lo
  g0[3] = (unsigned)((ga >> 32) & 0x01FFFFFFu) | (2u << 30);  // addr hi|type=2

  i32x8 g1;
  // dword0: data_size=2 (4B) @bits17:16, pad_enable @20, pad_interval=5
  // (64 dwords) @bits24:22, pad_amount=0 (1 dword) @bits31:25
  g1[0] = (2 << 16) | (1 << 20) | (5 << 22);
  g1[1] = (int)(((unsigned)N_DIM & 0xFFFFu) << 16);     // tensor_dim0[15:0]
  g1[2] = (int)((((unsigned)N_DIM >> 16) & 0xFFFFu) |   // tensor_dim0[31:16]
                ((unsigned)F_DIM << 16));               // tensor_dim1[15:0]
  g1[3] = (int)(64u << 16);                             // tile_dim0=64
  g1[4] = 64;                                           // tile_dim1=64
  g1[5] = (int)(unsigned)N_DIM;                         // dim0_stride lo
  g1[6] = 0;                                            // dim0_stride hi
  g1[7] = 0;

  i32x4 z4 = {};
#if __clang_major__ >= 23
  i32x8 z8 = {};
  __builtin_amdgcn_tensor_load_to_lds(g0, g1, z4, z4, z8, 0);
#else
  __builtin_amdgcn_tensor_load_to_lds(g0, g1, z4, z4, 0);
#endif
}

// ---------------------------------------------------------------------------
// 1. gather: x (NCHW) -> im [F, N]   (blocks-to-cols permutation)
// ---------------------------------------------------------------------------
__global__ void gather_kernel(const float* __restrict__ x,
                              float* __restrict__ im) {
  int idx = blockIdx.x * blockDim.x + threadIdx.x;
  if (idx >= TOTAL) return;
  int w  = idx % W_DIM;
  int h  = (idx / W_DIM) % H_DIM;
  int ch = (idx / (W_DIM * H_DIM)) % C_DIM;
  int b  = idx / (W_DIM * H_DIM * C_DIM);
  int rB = h >> 2, br = h & 3, cB = w >> 2, bc = w & 3;
  int f = (ch * MB + br) * MB + bc;
  int n = (rB * (W_DIM / MB) + cB) * B_DIM + b;
  im[(size_t)f * N_DIM + n] = x[idx];
}

// ---------------------------------------------------------------------------
// 2. per-row mean over N (one workgroup per feature row)
// ---------------------------------------------------------------------------
__global__ void mean_kernel(const float* __restrict__ im,
                            float* __restrict__ mn) {
  __shared__ float red[256];
  int f = blockIdx.x;
  const float* row = im + (size_t)f * N_DIM;
  float s = 0.f;
  for (int n = threadIdx.x; n < N_DIM; n += 256) s += row[n];
  red[threadIdx.x] = s;
  __syncthreads();
  for (int off = 128; off > 0; off >>= 1) {
    if (threadIdx.x < off) red[threadIdx.x] += red[threadIdx.x + off];
    __syncthreads();
  }
  if (threadIdx.x == 0) mn[f] = red[0] * (1.0f / (float)N_DIM);
}

// ---------------------------------------------------------------------------
// 3. center rows in place
// ---------------------------------------------------------------------------
__global__ void center_kernel(float* __restrict__ im,
                              const float* __restrict__ mn) {
  int idx = blockIdx.x * blockDim.x + threadIdx.x;
  if (idx >= TOTAL) return;
  int f = idx >> 16;            // N_DIM == 2^16
  im[idx] -= mn[f];
}

// ---------------------------------------------------------------------------
// 4a. zero cov (atomic-reduction target)
// ---------------------------------------------------------------------------
__global__ void zero_cov_kernel(float* __restrict__ cov) {
  int idx = blockIdx.x * blockDim.x + threadIdx.x;
  if (idx < F_DIM * F_DIM) cov[idx] = 0.0f;
}

// ---------------------------------------------------------------------------
// 4b. cov [F,F] += im_slice * im_slice^T / N
//     Workgroup = 256 threads (8 waves) owns a 64x64 output tile and one of
//     KSLICES K-ranges. A/B 64x64 tiles DMA'd into LDS by the TDM (padded to
//     65-float rows); each wave computes two 16x16 WMMA accumulators sharing
//     the A fragment. Epilogue reduces across K-slices with f32 atomics.
// ---------------------------------------------------------------------------
__global__ void cov_wmma_kernel(const float* __restrict__ im,
                                float* __restrict__ cov) {
  __shared__ float As[64 * 65];
  __shared__ float Bs[64 * 65];

  int wave = threadIdx.x >> 5;
  int lane = threadIdx.x & 31;
  int half = lane >> 4, lr = lane & 15;

  int tile  = blockIdx.x % ((F_DIM / 64) * (F_DIM / 64));   // 0..63
  int slice = blockIdx.x / ((F_DIM / 64) * (F_DIM / 64));   // 0..KSLICES-1
  int tm = tile / (F_DIM / 64), tn = tile % (F_DIM / 64);
  int m0 = tm * 64, n0 = tn * 64;
  int kBeg = slice * (N_DIM / KSLICES);
  int kEnd = kBeg + (N_DIM / KSLICES);

  int sr  = wave >> 1;            // sub-row 0..3 (16 rows each)
  int sc0 = (wave & 1) * 2;       // sub-cols sc0, sc0+1

  v8f acc0 = {}, acc1 = {};
  for (int k0 = kBeg; k0 < kEnd; k0 += 64) {
    __syncthreads();              // previous chunk fully consumed
    if (wave == 0) {
      tdm_load_tile64x64(im + (size_t)m0 * N_DIM + k0,
                         (unsigned)(uintptr_t)As);
      tdm_load_tile64x64(im + (size_t)n0 * N_DIM + k0,
                         (unsigned)(uintptr_t)Bs);
      __builtin_amdgcn_s_wait_tensorcnt(0);
    }
    __syncthreads();              // tiles visible to all waves

    const float* ar  = As + (sr * 16 + lr) * 65;
    const float* br0 = Bs + (sc0 * 16 + lr) * 65;
    const float* br1 = br0 + 16 * 65;
#pragma unroll
    for (int kk = 0; kk < 64; kk += 4) {
      v2f a, b0, b1;
      a.x  = ar[kk + 2 * half];      a.y  = ar[kk + 2 * half + 1];
      b0.x = br0[kk + 2 * half];     b0.y = br0[kk + 2 * half + 1];
      b1.x = br1[kk + 2 * half];     b1.y = br1[kk + 2 * half + 1];
      acc0 = wmma_f32_k4(a, b0, acc0);
      acc1 = wmma_f32_k4(a, b1, acc1);
    }
  }

  const float invN = 1.0f / (float)N_DIM;
#pragma unroll
  for (int v = 0; v < 8; ++v) {
    int m = m0 + sr * 16 + half * 8 + v;
    atomicAdd(&cov[(size_t)m * F_DIM + n0 + sc0 * 16 + lr], acc0[v] * invN);
    atomicAdd(&cov[(size_t)m * F_DIM + n0 + (sc0 + 1) * 16 + lr],
              acc1[v] * invN);
  }
}

// ---------------------------------------------------------------------------
// 5. U = I
// ---------------------------------------------------------------------------
__global__ void init_u_kernel(float* __restrict__ U) {
  int idx = blockIdx.x * blockDim.x + threadIdx.x;
  if (idx >= F_DIM * F_DIM) return;
  int r = idx / F_DIM, c = idx % F_DIM;
  U[idx] = (r == c) ? 1.0f : 0.0f;
}

// ---------------------------------------------------------------------------
// 6. Cyclic Jacobi eigensolver on the symmetric 512x512 cov.
// ---------------------------------------------------------------------------
__global__ void jacobi_kernel(float* __restrict__ A,
                              float* __restrict__ U,
                              float* __restrict__ stdd) {
  __shared__ int topS[256], botS[256];
  __shared__ float cS[256], sS[256];
  int tid = threadIdx.x;              // 0..511
  if (tid < 256) { topS[tid] = 2 * tid; botS[tid] = 2 * tid + 1; }
  __syncthreads();

  for (int sweep = 0; sweep < NSWEEPS; ++sweep) {
    for (int round = 0; round < F_DIM - 1; ++round) {
      if (tid < 256) {
        int p = topS[tid], q = botS[tid];
        if (p > q) { int t = p; p = q; q = t; }
        float app = A[(size_t)p * F_DIM + p];
        float aqq = A[(size_t)q * F_DIM + q];
        float apq = A[(size_t)p * F_DIM + q];
        float c = 1.0f, s = 0.0f;
        if (fabsf(apq) > 1e-20f) {
          float tau = (aqq - app) / (2.0f * apq);
          float t = copysignf(1.0f, tau) /
                    (fabsf(tau) + sqrtf(1.0f + tau * tau));
          c = 1.0f / sqrtf(1.0f + t * t);
          s = t * c;
        }
        cS[tid] = c; sS[tid] = s;
      }
      __syncthreads();

      // row rotations: A <- J^T A ; thread owns column j = tid
      for (int t = 0; t < 256; ++t) {
        int p = topS[t], q = botS[t];
        if (p > q) { int u = p; p = q; q = u; }
        float c = cS[t], s = sS[t];
        float ap = A[(size_t)p * F_DIM + tid];
        float aq = A[(size_t)q * F_DIM + tid];
        A[(size_t)p * F_DIM + tid] = c * ap + s * aq;
        A[(size_t)q * F_DIM + tid] = c * aq - s * ap;
      }
      __syncthreads();

      // column rotations: A <- A J, U <- U J ; thread owns row i = tid
      for (int t = 0; t < 256; ++t) {
        int p = topS[t], q = botS[t];
        if (p > q) { int u = p; p = q; q = u; }
        float c = cS[t], s = sS[t];
        float ap = A[(size_t)tid * F_DIM + p];
        float aq = A[(size_t)tid * F_DIM + q];
        A[(size_t)tid * F_DIM + p] = c * ap + s * aq;
        A[(size_t)tid * F_DIM + q] = c * aq - s * ap;
        float up = U[(size_t)tid * F_DIM + p];
        float uq = U[(size_t)tid * F_DIM + q];
        U[(size_t)tid * F_DIM + p] = c * up + s * uq;
        U[(size_t)tid * F_DIM + q] = c * uq - s * up;
      }
      __syncthreads();

      if (tid == 0) {   // rotate tournament pairing (player 0 fixed)
        int tl = topS[255];
        for (int i = 255; i >= 2; --i) topS[i] = topS[i - 1];
        topS[1] = botS[0];
        for (int i = 0; i < 255; ++i) botS[i] = botS[i + 1];
        botS[255] = tl;
      }
      __syncthreads();
    }
  }
  stdd[tid] = sqrtf(fmaxf(A[(size_t)tid * F_DIM + tid], 1e-12f));
}

// ---------------------------------------------------------------------------
// 7. proj = U^T * im, whiten by 1/sigma, 8-bit fake-quant, re-scale by sigma.
// ---------------------------------------------------------------------------
__global__ void proj_wmma_kernel(const float* __restrict__ U,
                                 const float* __restrict__ im,
                                 const float* __restrict__ stdd,
                                 const float* __restrict__ clampv,
                                 float* __restrict__ stage) {
  int wave = threadIdx.x >> 5;
  int lane = threadIdx.x & 31;
  int tile = blockIdx.x * (blockDim.x >> 5) + wave;
  int tm = tile % (F_DIM / 16);
  int tn = tile / (F_DIM / 16);
  int m0 = tm * 16, n0 = tn * 16;
  int half = lane >> 4, lr = lane & 15;

  v8f acc = {};
#pragma unroll 8
  for (int k = 0; k < F_DIM; k += 4) {
    v2f a, b;
    a.x = U[(size_t)(k + 2 * half + 0) * F_DIM + m0 + lr];   // A = U^T
    a.y = U[(size_t)(k + 2 * half + 1) * F_DIM + m0 + lr];
    b.x = im[(size_t)(k + 2 * half + 0) * N_DIM + n0 + lr];
    b.y = im[(size_t)(k + 2 * half + 1) * N_DIM + n0 + lr];
    acc = wmma_f32_k4(a, b, acc);
  }

  float cval = clampv[0];
  float scale = 255.0f / (2.0f * cval);
  float invscale = (2.0f * cval) / 255.0f;
#pragma unroll
  for (int v = 0; v < 8; ++v) {
    int m = m0 + half * 8 + v;
    float sd = stdd[m];
    float p = acc[v] / sd;
    float r = (fminf(fmaxf(p, -cval), cval) + cval) * scale;
    float q = rintf(r) * invscale - cval;
    stage[(size_t)m * N_DIM + n0 + lr] = q * sd;
  }
}

// ---------------------------------------------------------------------------
// 8. rec = U * qstd + mn  ->  recbuf
// ---------------------------------------------------------------------------
__global__ void rec_wmma_kernel(const float* __restrict__ U,
                                const float* __restrict__ stage,
                                const float* __restrict__ mn,
                                float* __restrict__ recbuf) {
  int wave = threadIdx.x >> 5;
  int lane = threadIdx.x & 31;
  int tile = blockIdx.x * (blockDim.x >> 5) + wave;
  int tm = tile % (F_DIM / 16);
  int tn = tile / (F_DIM / 16);
  int m0 = tm * 16, n0 = tn * 16;
  int half = lane >> 4, lr = lane & 15;

  const float* arow = U + (size_t)(m0 + lr) * F_DIM;
  v8f acc = {};
#pragma unroll 8
  for (int k = 0; k < F_DIM; k += 4) {
    v2f a, b;
    a.x = arow[k + 2 * half + 0];
    a.y = arow[k + 2 * half + 1];
    b.x = stage[(size_t)(k + 2 * half + 0) * N_DIM + n0 + lr];
    b.y = stage[(size_t)(k + 2 * half + 1) * N_DIM + n0 + lr];
    acc = wmma_f32_k4(a, b, acc);
  }

#pragma unroll
  for (int v = 0; v < 8; ++v) {
    int m = m0 + half * 8 + v;
    recbuf[(size_t)m * N_DIM + n0 + lr] = acc[v] + mn[m];
  }
}

// ---------------------------------------------------------------------------
// 9. scatter: recbuf [F,N] -> out (NCHW, cols-to-blocks inverse permutation)
// ---------------------------------------------------------------------------
__global__ void scatter_kernel(const float* __restrict__ recbuf,
                               float* __restrict__ out) {
  int idx = blockIdx.x * blockDim.x + threadIdx.x;
  if (idx >= TOTAL) return;
  int w  = idx % W_DIM;
  int h  = (idx / W_DIM) % H_DIM;
  int ch = (idx / (W_DIM * H_DIM)) % C_DIM;
  int b  = idx / (W_DIM * H_DIM * C_DIM);
  int rB = h >> 2, br = h & 3, cB = w >> 2, bc = w & 3;
  int f = (ch * MB + br) * MB + bc;
  int n = (rB * (W_DIM / MB) + cB) * B_DIM + b;
  out[idx] = recbuf[(size_t)f * N_DIM + n];
}

// ---------------------------------------------------------------------------
extern "C" void kernel_launch(void* const* d_in, const int* in_sizes, int n_in,
                              void* d_out, int out_size, void* d_ws,
                              size_t ws_size, hipStream_t stream) {
  const float* x      = (const float*)d_in[0];
  const float* clampv = (const float*)d_in[1];
  float* out = (float*)d_out;

  char* ws = (char*)d_ws;
  float* im   = (float*)ws;                                   // F*N (134 MB)
  size_t off  = (size_t)F_DIM * N_DIM * sizeof(float);
  float* cov  = (float*)(ws + off);  off += (size_t)F_DIM * F_DIM * sizeof(float);
  float* U    = (float*)(ws + off);  off += (size_t)F_DIM * F_DIM * sizeof(float);
  float* stdd = (float*)(ws + off);  off += F_DIM * sizeof(float);
  float* mn   = (float*)(ws + off);

  float* stage  = out;   // qstd staging lives in d_out (same element count)
  float* recbuf = im;    // im is dead after proj; reuse for reconstruction

  const int elemBlocks = (TOTAL + 255) / 256;                  // 131072
  const int covBlocks  = (F_DIM / 64) * (F_DIM / 64) * KSLICES; // 1024
  const int gemmBlocks = (F_DIM / 16) * (N_DIM / 16) / 8;      // 16384

  gather_kernel<<<elemBlocks, 256, 0, stream>>>(x, im);
  mean_kernel<<<F_DIM, 256, 0, stream>>>(im, mn);
  center_kernel<<<elemBlocks, 256, 0, stream>>>(im, mn);
  zero_cov_kernel<<<(F_DIM * F_DIM + 255) / 256, 256, 0, stream>>>(cov);
  cov_wmma_kernel<<<covBlocks, 256, 0, stream>>>(im, cov);
  init_u_kernel<<<(F_DIM * F_DIM + 255) / 256, 256, 0, stream>>>(U);
  jacobi_kernel<<<1, F_DIM, 0, stream>>>(cov, U, stdd);
  proj_wmma_kernel<<<gemmBlocks, 256, 0, stream>>>(U, im, stdd, clampv, stage);
  rec_wmma_kernel<<<gemmBlocks, 256, 0, stream>>>(U, stage, mn, recbuf);
  scatter_kernel<<<elemBlocks, 256, 0, stream>>>(recbuf, out);
}